// QFLayer_8607114461283
// MI455X (gfx1250) — compile-verified
//
#include <hip/hip_runtime.h>
#include <math.h>

// ---------------------------------------------------------------------------
// QFormer layer, algebraically collapsed:
//   cross-attn K/V rows identical => softmax uniform => x = im_repr@ca_wv+ca_bv
//   h = gelu(x@fi_w1+b1); y = x + h@fi_w2+b2; out = LayerNorm(y)*g+b
// One fused kernel: 16 blocks x 256 threads (8 wave32), 16 batch rows/block.
// GEMMs via V_WMMA_F32_16X16X4_F32; im tile staged via global_load_async_to_lds;
// K-outer / tiles-inner loops give 4-6 independent WMMA accumulate chains/wave.
// ---------------------------------------------------------------------------

typedef __attribute__((ext_vector_type(2))) float v2f;
typedef __attribute__((ext_vector_type(8))) float v8f;

#define NBATCH 256
#define DD     768   // hidden
#define ENC    128   // encoder dim
#define FI     256   // intermediate dim
#define MT     16    // rows per block (WMMA M)
#define XS     772   // LDS stride for x/y tile (772 % 64 == 4 -> conflict-free)
#define HS     260   // LDS stride for h tile   (260 % 64 == 4)
#define IMS    132   // LDS stride for im tile  (132 % 64 == 4)

#define WMMA_F32(a, b, c) \
    __builtin_amdgcn_wmma_f32_16x16x4_f32(false, (a), false, (b), (short)0, (c), false, false)

__global__ __launch_bounds__(256) void qf_fused_kernel(
    const float* __restrict__ im,    // [256,128]  im_repr
    const float* __restrict__ wv,    // [128,768]  ca_wv
    const float* __restrict__ bv,    // [768]      ca_bv
    const float* __restrict__ w1,    // [768,256]  fi_w1
    const float* __restrict__ b1,    // [256]      fi_b1
    const float* __restrict__ w2,    // [256,768]  fi_w2
    const float* __restrict__ b2,    // [768]      fi_b2
    const float* __restrict__ lng,   // [768]      ln_g
    const float* __restrict__ lnb,   // [768]      ln_b
    float* __restrict__ out)         // [256,768]
{
    __shared__ __align__(16) float sX[MT * XS];   // x, later y
    __shared__ __align__(16) float sH[MT * HS];   // h, later LN scratch
    __shared__ __align__(16) float sIm[MT * IMS]; // staged im tile

    const int tid     = threadIdx.x;
    const int wave    = tid >> 5;        // 0..7
    const int lane    = tid & 31;
    const int hi      = lane >> 4;       // lane half: K pair select / M+8
    const int l16     = lane & 15;
    const int rowBase = blockIdx.x * MT; // 16 batch rows per block

    // ---- Async-copy im tile (16x128 f32 = 8KB) into LDS (ASYNCcnt path). ----
    // Each thread moves 2 x 16B chunks; chunks never cross a 128-float row.
    {
        #pragma unroll
        for (int it = 0; it < 2; ++it) {
            const int g    = (tid + it * 256) * 4;        // flat float index
            const int row  = g >> 7;                      // /128
            const int col  = g & 127;
            const uint32_t ldsa = (uint32_t)(uintptr_t)(sIm + row * IMS + col);
            const uint64_t ga   = (uint64_t)(uintptr_t)(im + (size_t)(rowBase + row) * ENC + col);
            asm volatile("global_load_async_to_lds_b128 %0, %1, off"
                         :: "v"(ldsa), "v"(ga) : "memory");
        }
    }

    // Warm caches for later-stage weights (global_prefetch_b8).
    __builtin_prefetch(w1 + (size_t)tid * 768, 0, 3);
    __builtin_prefetch(w2 + (size_t)tid * 768, 0, 3);

    asm volatile("s_wait_asynccnt 0x0" ::: "memory");
    __syncthreads();

    // ---------------- Stage A: x[16,768] = im @ wv[128,768] + bv -------------
    // 48 N-tiles; 6 per wave, all 6 interleaved in the K loop (6 WMMA chains).
    {
        const float* ArowL = sIm + l16 * IMS;
        v8f acc[6] = {};
        for (int k0 = 0; k0 < ENC; k0 += 4) {
            const int ka = k0 + 2 * hi;                   // A/B: k = ka, ka+1
            const v2f a = *(const v2f*)(ArowL + ka);
            #pragma unroll
            for (int t = 0; t < 6; ++t) {
                const int ncol = (wave * 6 + t) * 16 + l16;
                v2f b;
                b.x = wv[(size_t)ka * DD + ncol];
                b.y = wv[(size_t)(ka + 1) * DD + ncol];
                acc[t] = WMMA_F32(a, b, acc[t]);
            }
        }
        #pragma unroll
        for (int t = 0; t < 6; ++t) {
            const int ncol = (wave * 6 + t) * 16 + l16;
            const float bias = bv[ncol];
            #pragma unroll
            for (int r = 0; r < 8; ++r) {
                const int m = r + 8 * hi;                 // C/D layout
                sX[m * XS + ncol] = acc[t][r] + bias;
            }
        }
    }
    __syncthreads();

    // ---------------- Stage B: h[16,256] = gelu(x @ w1[768,256] + b1) --------
    // 2 tiles/wave; K split into 2 interleaved accumulators => 4 WMMA chains.
    {
        const float* ArowX = sX + l16 * XS;
        v8f acc[2][2] = {};
        for (int k0 = 0; k0 < DD; k0 += 8) {
            const int ka0 = k0 + 2 * hi;
            const int ka1 = k0 + 4 + 2 * hi;
            const v2f a0 = *(const v2f*)(ArowX + ka0);
            const v2f a1 = *(const v2f*)(ArowX + ka1);
            #pragma unroll
            for (int t = 0; t < 2; ++t) {
                const int ncol = (wave * 2 + t) * 16 + l16;
                v2f b0, b1;
                b0.x = w1[(size_t)ka0 * FI + ncol];
                b0.y = w1[(size_t)(ka0 + 1) * FI + ncol];
                b1.x = w1[(size_t)ka1 * FI + ncol];
                b1.y = w1[(size_t)(ka1 + 1) * FI + ncol];
                acc[t][0] = WMMA_F32(a0, b0, acc[t][0]);
                acc[t][1] = WMMA_F32(a1, b1, acc[t][1]);
            }
        }
        #pragma unroll
        for (int t = 0; t < 2; ++t) {
            const int ncol = (wave * 2 + t) * 16 + l16;
            const float bias = b1[ncol];
            #pragma unroll
            for (int r = 0; r < 8; ++r) {
                const int m = r + 8 * hi;
                float v = acc[t][0][r] + acc[t][1][r] + bias;
                // exact GELU (torch nn.GELU default): 0.5*v*(1+erf(v/sqrt(2)))
                v = 0.5f * v * (1.0f + erff(v * 0.70710678118654752f));
                sH[m * HS + ncol] = v;
            }
        }
    }
    __syncthreads();

    // ---------------- Stage C: y = x + h @ w2[256,768] + b2 (into sX) --------
    // 6 tiles/wave interleaved in K loop (6 WMMA chains).
    {
        const float* ArowH = sH + l16 * HS;
        v8f acc[6] = {};
        for (int k0 = 0; k0 < FI; k0 += 4) {
            const int ka = k0 + 2 * hi;
            const v2f a = *(const v2f*)(ArowH + ka);
            #pragma unroll
            for (int t = 0; t < 6; ++t) {
                const int ncol = (wave * 6 + t) * 16 + l16;
                v2f b;
                b.x = w2[(size_t)ka * DD + ncol];
                b.y = w2[(size_t)(ka + 1) * DD + ncol];
                acc[t] = WMMA_F32(a, b, acc[t]);
            }
        }
        #pragma unroll
        for (int t = 0; t < 6; ++t) {
            const int ncol = (wave * 6 + t) * 16 + l16;
            const float bias = b2[ncol];
            #pragma unroll
            for (int r = 0; r < 8; ++r) {
                const int m = r + 8 * hi;
                const float y = acc[t][r] + bias + sX[m * XS + ncol]; // residual
                sX[m * XS + ncol] = y;  // in place: each lane owns its element
            }
        }
    }
    __syncthreads();

    // ---------------- LayerNorm over last dim (768), write output ------------
    {
        float* red = sH;                  // reuse: [0..255]=sum, [256..511]=sumsq
        const int row = tid >> 4;         // 0..15
        const int c0  = tid & 15;
        float s = 0.0f, ss = 0.0f;
        #pragma unroll 4
        for (int j = 0; j < DD / 16; ++j) {
            const float v = sX[row * XS + c0 + j * 16];
            s += v; ss += v * v;
        }
        red[row * 16 + c0]       = s;
        red[256 + row * 16 + c0] = ss;
        __syncthreads();

        float mu = 0.0f, m2 = 0.0f;
        #pragma unroll
        for (int j = 0; j < 16; ++j) {
            mu += red[row * 16 + j];
            m2 += red[256 + row * 16 + j];
        }
        mu *= (1.0f / 768.0f);
        const float var = m2 * (1.0f / 768.0f) - mu * mu;   // jnp.var (ddof=0)
        const float inv = rsqrtf(var + 1e-12f);

        float* orow = out + (size_t)(rowBase + row) * DD;
        #pragma unroll 4
        for (int j = 0; j < DD / 16; ++j) {
            const int c = c0 + j * 16;
            const float v = sX[row * XS + c];
            orow[c] = (v - mu) * inv * lng[c] + lnb[c];
        }
    }
}

extern "C" void kernel_launch(void* const* d_in, const int* in_sizes, int n_in,
                              void* d_out, int out_size, void* d_ws, size_t ws_size,
                              hipStream_t stream) {
    (void)in_sizes; (void)n_in; (void)out_size; (void)d_ws; (void)ws_size;
    // setup_inputs() order:
    //  0 im_repr, 1 q_repr, 2..8 mha_* (dead), 9 ca_wq, 10 ca_bq,
    //  11 ca_wk, 12 ca_bk, 13 ca_wv, 14 ca_bv, 15 ca_de,
    //  16 fi_w1, 17 fi_b1, 18 fi_w2, 19 fi_b2, 20 ln_g, 21 ln_b
    const float* im  = (const float*)d_in[0];
    const float* wv  = (const float*)d_in[13];
    const float* bv  = (const float*)d_in[14];
    const float* w1  = (const float*)d_in[16];
    const float* b1  = (const float*)d_in[17];
    const float* w2  = (const float*)d_in[18];
    const float* b2  = (const float*)d_in[19];
    const float* g   = (const float*)d_in[20];
    const float* b   = (const float*)d_in[21];
    float* out = (float*)d_out;

    hipLaunchKernelGGL(qf_fused_kernel, dim3(NBATCH / MT), dim3(256), 0, stream,
                       im, wv, bv, w1, b1, w2, b2, g, b, out);
}